// GGAHR2HK_52536039965311
// MI455X (gfx1250) — compile-verified
//
#include <hip/hip_runtime.h>

// ---------------------------------------------------------------------------
// HR2HK for MI455X (gfx1250, wave32).
// Output-bandwidth-bound problem: 382 MB of complex64 H(k) output vs ~72 MFLOP
// of rotation GEMMs. Strategy: wide nontemporal zero-fill, WMMA-rotated sparse
// block scatter with native no-return f32 atomics (global_atomic_add_f32).
// ---------------------------------------------------------------------------

#define PPAIR 58
#define SDIM  18          // spin-doubled block size
#define PAD   32          // zero-padded matrix dim for 16x16 WMMA tiling
#define TWO_PI_F 6.283185307179586f

typedef __attribute__((ext_vector_type(2))) float v2f;
typedef __attribute__((ext_vector_type(4))) float f4;
typedef __attribute__((ext_vector_type(8))) float v8f;

// Native f32 atomic add (no return value -> STOREcnt path, no CAS loop).
__device__ __forceinline__ void gadd(float* p, float v) {
  unsafeAtomicAdd(p, v);
}

// Map flat orbpair feature index p (0..57) -> (row, col, factor) in the 9x9
// upper-triangular assembly. FLIST = [1,3,5], offs = [0,1,4].
__device__ __forceinline__ void pmap(int p, int& r, int& c, float& f) {
  const int offs[3] = {0, 1, 4};
  const int sz[3]   = {1, 3, 5};
  int base = 0;
  for (int i = 0; i < 3; ++i)
    for (int j = i; j < 3; ++j) {
      const int cnt = sz[i] * sz[j];
      if (p < base + cnt) {
        const int q = p - base;
        r = offs[i] + q / sz[j];
        c = offs[j] + q % sz[j];
        f = (i == j) ? 0.5f : 1.0f;
        return;
      }
      base += cnt;
    }
  r = 0; c = 0; f = 0.0f;
}

// C = A * B (or A * B^T) for 32x32 zero-padded f32 matrices in LDS.
// Single wave32, EXEC all ones. Uses V_WMMA_F32_16X16X4_F32.
// Fragment layouts per CDNA5 ISA 7.12.2:
//   A 16x4 : VGPR0 = {K=k0 (lanes 0-15), K=k0+2 (lanes 16-31)}, VGPR1 = k0+1/k0+3
//   B 4x16 : VGPR0 = {row k0 (lanes 0-15), row k0+2 (lanes 16-31)}, VGPR1 = +1
//   C/D    : VGPR r = row (r + 0/8), N = lane%16
__device__ __forceinline__ void gemm32_wmma(const float* __restrict__ A,
                                            const float* __restrict__ B,
                                            float* __restrict__ C,
                                            bool transB, int lane) {
  const int l15    = lane & 15;
  const int khalf  = (lane >> 4) << 1;      // 0 or 2
  const int rowOff = (lane < 16) ? 0 : 8;
#pragma unroll
  for (int mi = 0; mi < 2; ++mi) {
#pragma unroll
    for (int ni = 0; ni < 2; ++ni) {
      v8f acc = {0.f, 0.f, 0.f, 0.f, 0.f, 0.f, 0.f, 0.f};
#pragma unroll
      for (int kk = 0; kk < 5; ++kk) {      // K = 0..19 covers 18 valid + pad
        const int kb = kk * 4 + khalf;
        v2f a, b;
        a.x = A[(mi * 16 + l15) * PAD + kb];
        a.y = A[(mi * 16 + l15) * PAD + kb + 1];
        const int n = ni * 16 + l15;
        if (!transB) { b.x = B[kb * PAD + n];  b.y = B[(kb + 1) * PAD + n]; }
        else         { b.x = B[n * PAD + kb];  b.y = B[n * PAD + kb + 1];   }
        acc = __builtin_amdgcn_wmma_f32_16x16x4_f32(
            /*neg_a=*/false, a, /*neg_b=*/false, b,
            /*c_mod=*/(short)0, acc, /*reuse_a=*/false, /*reuse_b=*/false);
      }
#pragma unroll
      for (int r = 0; r < 8; ++r)
        C[(mi * 16 + rowOff + r) * PAD + ni * 16 + l15] = acc[r];
    }
  }
}

// ---------------------------------------------------------------------------
// Kernel 1: zero-fill output with wide nontemporal stores (streaming, 382 MB).
// ---------------------------------------------------------------------------
__global__ void hr2hk_zero(f4* __restrict__ out, long long n4) {
  long long i = (long long)blockIdx.x * blockDim.x + threadIdx.x;
  const long long stride = (long long)gridDim.x * blockDim.x;
  const f4 z = {0.f, 0.f, 0.f, 0.f};
  for (; i < n4; i += stride)
    __builtin_nontemporal_store(z, out + i);
}

// ---------------------------------------------------------------------------
// Kernel 2: onsite blocks. One wave32 per atom.
// T = R_a * spin_double(assemble(onsite_a)) * R_a^T  (real)
// out[k, a*18+i, a*18+j].re += T[i][j] + T[j][i]   for every k
// ---------------------------------------------------------------------------
__global__ __launch_bounds__(32) void hr2hk_onsite(
    const float* __restrict__ ons, const float* __restrict__ R,
    float* __restrict__ out, int nA, int nK) {
  __shared__ float Ha[PAD * PAD];
  __shared__ float Ra[PAD * PAD];
  __shared__ float T1[PAD * PAD];
  const int a = blockIdx.x;
  const int lane = threadIdx.x;

  for (int t = lane; t < PAD * PAD; t += 32) { Ha[t] = 0.f; Ra[t] = 0.f; }
  for (int t = lane; t < SDIM * SDIM; t += 32)
    Ra[(t / SDIM) * PAD + (t % SDIM)] = R[(size_t)a * SDIM * SDIM + t];
  for (int p = lane; p < PPAIR; p += 32) {
    int r, c; float f;
    pmap(p, r, c, f);
    const float v = f * ons[(size_t)a * PPAIR + p];
    Ha[(2 * r) * PAD + (2 * c)] = v;           // spin 0
    Ha[(2 * r + 1) * PAD + (2 * c + 1)] = v;   // spin 1 (kron with I2)
  }
  __syncthreads();

  gemm32_wmma(Ra, Ha, T1, false, lane);  // T1 = R * H
  __syncthreads();
  gemm32_wmma(T1, Ra, Ha, true, lane);   // Ha = T1 * R^T  (reuse as T2)
  __syncthreads();

  const size_t NS = (size_t)nA * SDIM;
  const int Kc = nK > 8 ? 8 : nK;
  for (int t = lane; t < SDIM * SDIM; t += 32) {
    const int i = t / SDIM, j = t % SDIM;
    const float v = Ha[i * PAD + j] + Ha[j * PAD + i];  // H + H^T (real)
    const size_t r1 = (size_t)a * SDIM + i;
    const size_t c1 = (size_t)a * SDIM + j;
    for (int k = 0; k < Kc; ++k) {
      float* p0 = out + ((((size_t)k * NS + r1) * NS + c1) << 1);
      gadd(p0, v);                       // imaginary part stays 0
    }
  }
}

// ---------------------------------------------------------------------------
// Kernel 3: edge (hopping) blocks. One wave32 per edge.
// T = R_src * spin_double(assemble(hop_e)) * R_dst^T  (real)
// For each k: phase = exp(-2*pi*i k.S)
//   out[k, src*18+i, dst*18+j] += T * phase
//   out[k, dst*18+j, src*18+i] += conj(T * phase)     (hermitianization)
// ---------------------------------------------------------------------------
__global__ __launch_bounds__(32) void hr2hk_edge(
    const float* __restrict__ hop, const float* __restrict__ R,
    const float* __restrict__ kp, const float* __restrict__ sh,
    const int* __restrict__ eidx, float* __restrict__ out,
    int nA, int nE, int nK) {
  __shared__ float Ha[PAD * PAD];
  __shared__ float Ra[PAD * PAD];
  __shared__ float Rb[PAD * PAD];
  __shared__ float T1[PAD * PAD];

  const int e = blockIdx.x;
  const int lane = threadIdx.x;
  const int src = eidx[e];
  const int dst = eidx[nE + e];

  for (int t = lane; t < PAD * PAD; t += 32) { Ha[t] = 0.f; Ra[t] = 0.f; Rb[t] = 0.f; }
  for (int t = lane; t < SDIM * SDIM; t += 32) {
    Ra[(t / SDIM) * PAD + (t % SDIM)] = R[(size_t)src * SDIM * SDIM + t];
    Rb[(t / SDIM) * PAD + (t % SDIM)] = R[(size_t)dst * SDIM * SDIM + t];
  }
  for (int p = lane; p < PPAIR; p += 32) {
    int r, c; float f;
    pmap(p, r, c, f);
    const float v = f * hop[(size_t)e * PPAIR + p];
    Ha[(2 * r) * PAD + (2 * c)] = v;
    Ha[(2 * r + 1) * PAD + (2 * c + 1)] = v;
  }
  __syncthreads();

  gemm32_wmma(Ra, Ha, T1, false, lane);  // T1 = R_src * H
  __syncthreads();
  gemm32_wmma(T1, Rb, Ha, true, lane);   // Ha = T1 * R_dst^T
  __syncthreads();

  // Bloch phases: theta = 2*pi*(k . S); phase = cos(theta) - i sin(theta)
  float ck[8], sk[8];
  const int Kc = nK > 8 ? 8 : nK;
  const float Sx = sh[3 * e], Sy = sh[3 * e + 1], Sz = sh[3 * e + 2];
  for (int k = 0; k < Kc; ++k) {
    const float th = TWO_PI_F * (kp[3 * k] * Sx + kp[3 * k + 1] * Sy + kp[3 * k + 2] * Sz);
    __sincosf(th, &sk[k], &ck[k]);
  }

  const size_t NS = (size_t)nA * SDIM;
  for (int t = lane; t < SDIM * SDIM; t += 32) {
    const int i = t / SDIM, j = t % SDIM;
    const float v = Ha[i * PAD + j];
    const size_t r1 = (size_t)src * SDIM + i;
    const size_t c1 = (size_t)dst * SDIM + j;
    for (int k = 0; k < Kc; ++k) {
      const float re = v * ck[k];
      const float im = -v * sk[k];
      float* p0 = out + ((((size_t)k * NS + r1) * NS + c1) << 1);
      gadd(p0, re);
      gadd(p0 + 1, im);
      float* p1 = out + ((((size_t)k * NS + c1) * NS + r1) << 1);
      gadd(p1, re);
      gadd(p1 + 1, -im);
    }
  }
}

// ---------------------------------------------------------------------------
extern "C" void kernel_launch(void* const* d_in, const int* in_sizes, int n_in,
                              void* d_out, int out_size, void* d_ws, size_t ws_size,
                              hipStream_t stream) {
  const float* hop = (const float*)d_in[0];   // [E, 58]
  const float* ons = (const float*)d_in[1];   // [A, 58]
  const float* R   = (const float*)d_in[2];   // [A, 18, 18]
  const float* kp  = (const float*)d_in[3];   // [K, 3]
  const float* sh  = (const float*)d_in[4];   // [E, 3]
  const int*  eidx = (const int*)d_in[5];     // [2, E]
  float* out = (float*)d_out;                 // complex64 [K, A*18, A*18] as re/im pairs

  const int E = in_sizes[0] / PPAIR;
  const int A = in_sizes[1] / PPAIR;
  const int K = in_sizes[3] / 3;

  const long long n4 = (long long)out_size / 4;
  hr2hk_zero<<<16384, 256, 0, stream>>>((f4*)out, n4);
  hr2hk_onsite<<<A, 32, 0, stream>>>(ons, R, out, A, K);
  hr2hk_edge<<<E, 32, 0, stream>>>(hop, R, kp, sh, eidx, out, A, E, K);
}